// GVF_MoE_V4_Model_4002909520310
// MI455X (gfx1250) — compile-verified
//
#include <hip/hip_runtime.h>
#include <math.h>

// ---------------------------------------------------------------------------
// GVF MoE V4: GCN (2 layers, edge scatter) + context MLP + softmax gate.
// MI455X strategy:
//  - all per-node dense matmuls via V_WMMA_F32_16X16X4_F32 (exact fp32)
//  - branch-free fragment loads (clamped addresses + value selects)
//  - result tiles staged in LDS, row-scaled at store time, coalesced b128
//  - edge scatters: f32 atomics into L2-resident tables (~20MB << 192MB L2)
// ---------------------------------------------------------------------------

typedef float v2f __attribute__((ext_vector_type(2)));
typedef float v8f __attribute__((ext_vector_type(8)));

#define WPB 8   // waves per 256-thread block (wave32)

static __device__ __forceinline__ v8f wmma_f32(v2f a, v2f b, v8f c) {
  // D = A(16x4,f32) * B(4x16,f32) + C(16x16,f32)
  return __builtin_amdgcn_wmma_f32_16x16x4_f32(false, a, false, b,
                                               (short)0, c, false, false);
}

// B fragment from row-major K x Ncols weights, zero-padded to 4x16.
// Branch-free: unconditional load from clamped index, then value select.
static __device__ __forceinline__ v2f b_frag(const float* __restrict__ W,
                                             int lane, int k0, int K, int Ncols) {
  int n  = lane & 15;
  int kb = k0 + ((lane >> 4) << 1);
  int nm  = (n      < Ncols) ? n      : (Ncols - 1);
  int kc0 = (kb     < K)     ? kb     : (K - 1);
  int kc1 = (kb + 1 < K)     ? kb + 1 : (K - 1);
  float w0 = W[kc0 * Ncols + nm];
  float w1 = W[kc1 * Ncols + nm];
  v2f b;
  b.x = (kb     < K && n < Ncols) ? w0 : 0.0f;
  b.y = (kb + 1 < K && n < Ncols) ? w1 : 0.0f;
  return b;
}

// A fragment from x (N x 16 row-major), feature window [co, co+K).
static __device__ __forceinline__ v2f a_frag_x(const float* __restrict__ x,
                                               int N, int m0, int lane,
                                               int k0, int co, int K) {
  int r = m0 + (lane & 15);
  if (r >= N) r = N - 1;
  int kb = k0 + ((lane >> 4) << 1);
  const float* rowp = x + (size_t)r * 16;
  int i0 = co + kb;     i0 = (i0 < 16) ? i0 : 15;
  int i1 = co + kb + 1; i1 = (i1 < 16) ? i1 : 15;
  float a0 = rowp[i0];
  float a1 = rowp[i1];
  v2f a;
  a.x = (kb     < K) ? a0 : 0.0f;
  a.y = (kb + 1 < K) ? a1 : 0.0f;
  return a;
}

// A fragment from a flat 16x16 LDS tile (row stride 16).
static __device__ __forceinline__ v2f a_frag_lds(const float* t, int lane, int k0) {
  int r  = lane & 15;
  int kb = k0 + ((lane >> 4) << 1);
  v2f a;
  a.x = t[r * 16 + kb];
  a.y = t[r * 16 + kb + 1];
  return a;
}

// ---------------------------------------------------------------------------
// Kernel 1: init deg=1 (self-loops) and zero edge accumulators
// ---------------------------------------------------------------------------
__global__ void k_init(float* __restrict__ deg, float* __restrict__ acc1,
                       float* __restrict__ acc2, int N) {
  int i = blockIdx.x * blockDim.x + threadIdx.x;
  if (i >= N) return;
  deg[i] = 1.0f;
  float4 z = make_float4(0.f, 0.f, 0.f, 0.f);
  float4* a1 = (float4*)(acc1 + (size_t)i * 16);
  a1[0] = z; a1[1] = z; a1[2] = z; a1[3] = z;
  float4* a2 = (float4*)(acc2 + (size_t)i * 8);
  a2[0] = z; a2[1] = z;
}

// ---------------------------------------------------------------------------
// Kernel 2: degree count (atomics into L2-resident 400KB table)
// ---------------------------------------------------------------------------
__global__ void k_degree(const int* __restrict__ col, float* __restrict__ deg,
                         int NE) {
  int e = blockIdx.x * blockDim.x + threadIdx.x;
  if (e < NE) atomicAdd(&deg[col[e]], 1.0f);
}

// ---------------------------------------------------------------------------
// Kernel 3: dinv = rsqrt(deg);  xw1s = dinv * (x[:, :3] @ gW1)   [WMMA]
// ---------------------------------------------------------------------------
__global__ void __launch_bounds__(256)
k_dense1(const float* __restrict__ x, const float* __restrict__ gW1,
         const float* __restrict__ deg, float* __restrict__ dinv,
         float* __restrict__ xw1s, int N) {
  __shared__ float ot[WPB][256];
  int wslot = threadIdx.x >> 5;
  int lane  = threadIdx.x & 31;
  int m0 = (blockIdx.x * WPB + wslot) * 16;
  if (m0 >= N) return;                         // wave-uniform
  bool full = (m0 + 16) <= N;
  float* t = ot[wslot];

  // every lane: dinv of row (lane & 15); deg >= 1 so raw v_rsq is fine
  int rr  = m0 + (lane & 15);
  int rrc = (rr < N) ? rr : N - 1;
  float d = __builtin_amdgcn_rsqf(deg[rrc]);

  v8f c = {};
  v2f a = a_frag_x(x, N, m0, lane, 0, /*co=*/0, /*K=*/3);
  v2f b = b_frag(gW1, lane, 0, 3, 16);
  c = wmma_f32(a, b, c);

  if (lane < 16 && rr < N) dinv[rr] = d;

  int n  = lane & 15;
  int mo = (lane >> 4) << 3;
  for (int v = 0; v < 8; ++v)
    t[(mo + v) * 16 + n] = c[v];               // unscaled; scale at store
  __builtin_amdgcn_wave_barrier();

  if (full) {
    // chunk 'lane' covers row lane>>2, chunk 'lane+32' covers row 8+(lane>>2)
    float dv0 = __shfl(d, lane >> 2, 32);
    float dv1 = __shfl(d, 8 + (lane >> 2), 32);
    const float4* src = (const float4*)t;
    float4 v0 = src[lane];
    float4 v1 = src[lane + 32];
    v0.x *= dv0; v0.y *= dv0; v0.z *= dv0; v0.w *= dv0;
    v1.x *= dv1; v1.y *= dv1; v1.z *= dv1; v1.w *= dv1;
    float4* dst = (float4*)(xw1s + (size_t)m0 * 16);
    dst[lane]      = v0;
    dst[lane + 32] = v1;
  } else {
    for (int v = 0; v < 8; ++v) {
      int r = m0 + mo + v;
      float dv = __shfl(d, mo + v, 32);
      if (r < N) xw1s[(size_t)r * 16 + n] = t[(mo + v) * 16 + n] * dv;
    }
  }
}

// ---------------------------------------------------------------------------
// Kernel 4: scatter layer 1 — acc1[col] += xw1s[row]   (4 lanes per edge)
// ---------------------------------------------------------------------------
__global__ void k_scatter1(const int* __restrict__ row, const int* __restrict__ col,
                           const float* __restrict__ xw1s, float* __restrict__ acc1,
                           int NE) {
  int t = blockIdx.x * blockDim.x + threadIdx.x;
  int e = t >> 2;
  if (e >= NE) return;
  int q = t & 3;
  int r = row[e], cdst = col[e];
  float4 v = ((const float4*)(xw1s + (size_t)r * 16))[q];
  float* dst = acc1 + (size_t)cdst * 16 + q * 4;
  atomicAdd(dst + 0, v.x);
  atomicAdd(dst + 1, v.y);
  atomicAdd(dst + 2, v.z);
  atomicAdd(dst + 3, v.w);
}

// ---------------------------------------------------------------------------
// Kernel 5: h = relu(dinv*(acc1 + xw1s) + gb1);  xw2s = dinv*(h @ gW2) [WMMA]
// ---------------------------------------------------------------------------
__global__ void __launch_bounds__(256)
k_dense2(const float* __restrict__ acc1, const float* __restrict__ xw1s,
         const float* __restrict__ dinv, const float* __restrict__ gW2,
         const float* __restrict__ gb1, float* __restrict__ xw2s, int N) {
  __shared__ float ht[WPB][256];
  __shared__ float dv[WPB][16];
  int wslot = threadIdx.x >> 5;
  int lane  = threadIdx.x & 31;
  int m0 = (blockIdx.x * WPB + wslot) * 16;
  if (m0 >= N) return;                         // wave-uniform
  bool full = (m0 + 16) <= N;
  float* t   = ht[wslot];
  float* dvs = dv[wslot];

  // coalesced load of the 16x16 tile: 2 float4 chunks per lane
  for (int h = 0; h < 2; ++h) {
    int f   = lane + h * 32;                   // flat float4 chunk 0..63
    int row = f >> 2;
    int cc4 = f & 3;
    int r   = m0 + row;
    int rc  = (r < N) ? r : N - 1;
    float dvr = dinv[rc];
    if (cc4 == 0) dvs[row] = dvr;
    float4 a4 = *(const float4*)(acc1 + (size_t)rc * 16 + cc4 * 4);
    float4 s4 = *(const float4*)(xw1s + (size_t)rc * 16 + cc4 * 4);
    float4 b4 = ((const float4*)gb1)[cc4];
    float4 hv;
    hv.x = fmaxf(dvr * (a4.x + s4.x) + b4.x, 0.0f);
    hv.y = fmaxf(dvr * (a4.y + s4.y) + b4.y, 0.0f);
    hv.z = fmaxf(dvr * (a4.z + s4.z) + b4.z, 0.0f);
    hv.w = fmaxf(dvr * (a4.w + s4.w) + b4.w, 0.0f);
    *(float4*)&t[row * 16 + cc4 * 4] = hv;
  }
  __builtin_amdgcn_wave_barrier();

  v8f c = {};
  for (int k0 = 0; k0 < 16; k0 += 4) {
    v2f a = a_frag_lds(t, lane, k0);
    v2f b = b_frag(gW2, lane, k0, 16, 8);
    c = wmma_f32(a, b, c);
  }
  __builtin_amdgcn_wave_barrier();             // done reading t; reuse as out tile

  int n  = lane & 15;
  int mo = (lane >> 4) << 3;
  if (n < 8) {
    for (int v = 0; v < 8; ++v)
      t[(mo + v) * 8 + n] = c[v];              // unscaled; scale at store
  }
  __builtin_amdgcn_wave_barrier();

  if (full) {
    // chunk 'lane' covers row lane>>1 of the 16x8 tile
    float dvr = dvs[lane >> 1];
    float4 v4 = ((const float4*)t)[lane];
    v4.x *= dvr; v4.y *= dvr; v4.z *= dvr; v4.w *= dvr;
    ((float4*)(xw2s + (size_t)m0 * 8))[lane] = v4;
  } else if (n < 8) {
    for (int v = 0; v < 8; ++v) {
      int r = m0 + mo + v;
      if (r < N) xw2s[(size_t)r * 8 + n] = t[(mo + v) * 8 + n] * dvs[mo + v];
    }
  }
}

// ---------------------------------------------------------------------------
// Kernel 6: scatter layer 2 — acc2[col] += xw2s[row]   (2 lanes per edge)
// ---------------------------------------------------------------------------
__global__ void k_scatter2(const int* __restrict__ row, const int* __restrict__ col,
                           const float* __restrict__ xw2s, float* __restrict__ acc2,
                           int NE) {
  int t = blockIdx.x * blockDim.x + threadIdx.x;
  int e = t >> 1;
  if (e >= NE) return;
  int q = t & 1;
  int r = row[e], cdst = col[e];
  float4 v = ((const float4*)(xw2s + (size_t)r * 8))[q];
  float* dst = acc2 + (size_t)cdst * 8 + q * 4;
  atomicAdd(dst + 0, v.x);
  atomicAdd(dst + 1, v.y);
  atomicAdd(dst + 2, v.z);
  atomicAdd(dst + 3, v.w);
}

// ---------------------------------------------------------------------------
// Kernel 7: graph_out, context MLP, gate MLP + softmax, gated combine  [WMMA]
// ---------------------------------------------------------------------------
__global__ void __launch_bounds__(256)
k_final(const float* __restrict__ x, const float* __restrict__ acc2,
        const float* __restrict__ xw2s, const float* __restrict__ dinv,
        const float* __restrict__ gb2,
        const float* __restrict__ cW1, const float* __restrict__ cb1,
        const float* __restrict__ cW2, const float* __restrict__ cb2,
        const float* __restrict__ mW1, const float* __restrict__ mb1,
        const float* __restrict__ mW2, const float* __restrict__ mb2,
        float* __restrict__ out, int N) {
  __shared__ float ht[WPB][256];
  __shared__ float ct[WPB][128];
  __shared__ float g0s[WPB][16];
  int wslot = threadIdx.x >> 5;
  int lane  = threadIdx.x & 31;
  int m0 = (blockIdx.x * WPB + wslot) * 16;
  if (m0 >= N) return;                         // wave-uniform
  bool full = (m0 + 16) <= N;
  float* t  = ht[wslot];
  float* o  = ct[wslot];
  float* g0 = g0s[wslot];

  int n  = lane & 15;
  int mo = (lane >> 4) << 3;

  // graph_out in flat float4 layout: lane handles chunk (row=lane>>1, half=lane&1)
  int grow = lane >> 1, gcc = lane & 1;
  int gr  = m0 + grow;
  int grc = (gr < N) ? gr : N - 1;
  float  gdv = dinv[grc];
  float4 a24 = *(const float4*)(acc2 + (size_t)grc * 8 + gcc * 4);
  float4 s24 = *(const float4*)(xw2s + (size_t)grc * 8 + gcc * 4);
  float4 gb4 = ((const float4*)gb2)[gcc];
  float4 go4;
  go4.x = gdv * (a24.x + s24.x) + gb4.x;
  go4.y = gdv * (a24.y + s24.y) + gb4.y;
  go4.z = gdv * (a24.z + s24.z) + gb4.z;
  go4.w = gdv * (a24.w + s24.w) + gb4.w;

  // context hidden: relu(x[:,3:16] @ cW1 + cb1), K=13 padded to 16
  v8f ch = {};
  for (int k0 = 0; k0 < 16; k0 += 4) {
    v2f a = a_frag_x(x, N, m0, lane, k0, /*co=*/3, /*K=*/13);
    v2f b = b_frag(cW1, lane, k0, 13, 16);
    ch = wmma_f32(a, b, ch);
  }
  for (int v = 0; v < 8; ++v) t[(mo + v) * 16 + n] = fmaxf(ch[v] + cb1[n], 0.0f);
  __builtin_amdgcn_wave_barrier();

  // context out: hc @ cW2 (16 -> 8), stash (+cb2) into flat 16x8 tile
  v8f cc = {};
  for (int k0 = 0; k0 < 16; k0 += 4) {
    v2f a = a_frag_lds(t, lane, k0);
    v2f b = b_frag(cW2, lane, k0, 16, 8);
    cc = wmma_f32(a, b, cc);
  }
  if (n < 8) {
    for (int v = 0; v < 8; ++v) o[(mo + v) * 8 + n] = cc[v] + cb2[n];
  }
  __builtin_amdgcn_wave_barrier();             // t reads done; safe to overwrite

  // gate hidden: relu(x @ mW1 + mb1), K=16
  v8f cm = {};
  for (int k0 = 0; k0 < 16; k0 += 4) {
    v2f a = a_frag_x(x, N, m0, lane, k0, /*co=*/0, /*K=*/16);
    v2f b = b_frag(mW1, lane, k0, 16, 16);
    cm = wmma_f32(a, b, cm);
  }
  for (int v = 0; v < 8; ++v) t[(mo + v) * 16 + n] = fmaxf(cm[v] + mb1[n], 0.0f);
  __builtin_amdgcn_wave_barrier();

  // gate logits (16 -> 2) + 2-way softmax: one node per lane (lanes 0..15)
  if (lane < 16) {
    float l0 = mb2[0], l1 = mb2[1];
    for (int j = 0; j < 16; ++j) {
      float hv = t[lane * 16 + j];
      l0 += hv * mW2[j * 2 + 0];
      l1 += hv * mW2[j * 2 + 1];
    }
    float mx = fmaxf(l0, l1);
    float e0 = __expf(l0 - mx), e1 = __expf(l1 - mx);
    g0[lane] = e0 / (e0 + e1);
  }
  __builtin_amdgcn_wave_barrier();

  // combine in flat float4 space, coalesced b128 store
  float g  = g0[grow];
  float4 c4 = *(const float4*)&o[lane * 4];
  float4 r4;
  r4.x = g * go4.x + (1.0f - g) * c4.x;
  r4.y = g * go4.y + (1.0f - g) * c4.y;
  r4.z = g * go4.z + (1.0f - g) * c4.z;
  r4.w = g * go4.w + (1.0f - g) * c4.w;
  if (full) {
    ((float4*)(out + (size_t)m0 * 8))[lane] = r4;
  } else if (gr < N) {
    *(float4*)(out + (size_t)gr * 8 + gcc * 4) = r4;
  }
}

// ---------------------------------------------------------------------------
extern "C" void kernel_launch(void* const* d_in, const int* in_sizes, int n_in,
                              void* d_out, int out_size, void* d_ws, size_t ws_size,
                              hipStream_t stream) {
  const float* x   = (const float*)d_in[0];
  const int*   ei  = (const int*)  d_in[1];
  const float* gW1 = (const float*)d_in[2];
  const float* gb1 = (const float*)d_in[3];
  const float* gW2 = (const float*)d_in[4];
  const float* gb2 = (const float*)d_in[5];
  const float* cW1 = (const float*)d_in[6];
  const float* cb1 = (const float*)d_in[7];
  const float* cW2 = (const float*)d_in[8];
  const float* cb2 = (const float*)d_in[9];
  const float* mW1 = (const float*)d_in[10];
  const float* mb1 = (const float*)d_in[11];
  const float* mW2 = (const float*)d_in[12];
  const float* mb2 = (const float*)d_in[13];
  float* out = (float*)d_out;

  const int N  = in_sizes[0] / 16;   // 100000
  const int NE = in_sizes[1] / 2;    // 6400000
  const int* erow = ei;
  const int* ecol = ei + NE;

  // workspace carve (floats): deg | dinv | xw1s(16N) | acc1(16N) | xw2s(8N) | acc2(8N)
  float* deg  = (float*)d_ws;
  float* dinv = deg  + N;
  float* xw1s = dinv + N;
  float* acc1 = xw1s + (size_t)16 * N;
  float* xw2s = acc1 + (size_t)16 * N;
  float* acc2 = xw2s + (size_t)8 * N;

  const int tiles   = (N + 15) / 16;
  const int gBlocks = (tiles + WPB - 1) / WPB;

  k_init    <<<(N + 255) / 256, 256, 0, stream>>>(deg, acc1, acc2, N);
  k_degree  <<<(NE + 255) / 256, 256, 0, stream>>>(ecol, deg, NE);
  k_dense1  <<<gBlocks, 256, 0, stream>>>(x, gW1, deg, dinv, xw1s, N);
  k_scatter1<<<((size_t)NE * 4 + 255) / 256, 256, 0, stream>>>(erow, ecol, xw1s, acc1, NE);
  k_dense2  <<<gBlocks, 256, 0, stream>>>(acc1, xw1s, dinv, gW2, gb1, xw2s, N);
  k_scatter2<<<((size_t)NE * 2 + 255) / 256, 256, 0, stream>>>(erow, ecol, xw2s, acc2, NE);
  k_final   <<<gBlocks, 256, 0, stream>>>(x, acc2, xw2s, dinv, gb2,
                                          cW1, cb1, cW2, cb2,
                                          mW1, mb1, mW2, mb2, out, N);
}